// SoftPhysicsGNNLayer_12678743458332
// MI455X (gfx1250) — compile-verified
//
#include <hip/hip_runtime.h>
#include <hip/hip_bf16.h>

typedef __attribute__((ext_vector_type(2))) float v2f;
typedef __attribute__((ext_vector_type(4))) float v4f;
typedef __attribute__((ext_vector_type(8))) float v8f;

// ---------------------------------------------------------------------------
// Kernel 0: zero the net_current accumulator in workspace
// ---------------------------------------------------------------------------
__global__ void zero_kernel(float* __restrict__ p, int n) {
    int i = blockIdx.x * blockDim.x + threadIdx.x;
    if (i < n) p[i] = 0.0f;
}

// ---------------------------------------------------------------------------
// Kernel 1: per-edge physics + residual MLP + scatter-add (segment sums)
//   I_pred written to d_out edge section; net_current accumulated via
//   fp32 global atomics (L2-resident, 800KB).
//   edge_features is a 205MB one-shot stream -> non-temporal loads.
// ---------------------------------------------------------------------------
__global__ __launch_bounds__(256) void edge_kernel(
    const float* __restrict__ V_node,
    const int*   __restrict__ senders,
    const int*   __restrict__ receivers,
    const float* __restrict__ ef,      // (E,16)
    const float* __restrict__ W_res,   // (20,2) row-major
    const float* __restrict__ b_res,   // (2)
    float*       __restrict__ I_pred,  // (E,2)
    float*       __restrict__ net_cur, // (N,2)
    int n_edges)
{
    int e = blockIdx.x * blockDim.x + threadIdx.x;
    if (e >= n_edges) return;

    int s = senders[e];
    int r = receivers[e];

    // gather first two node-voltage components (L2-resident gather set)
    float2 vr = *(const float2*)(V_node + (size_t)r * 64);
    float2 vs = *(const float2*)(V_node + (size_t)s * 64);
    float Vre = vr.x - vs.x;
    float Vim = vr.y - vs.y;

    // streaming read of 16 edge features (non-temporal: no reuse)
    const v4f* efp = (const v4f*)(ef + (size_t)e * 16);
    v4f f[4];
    f[0] = __builtin_nontemporal_load(efp + 0);
    f[1] = __builtin_nontemporal_load(efp + 1);
    f[2] = __builtin_nontemporal_load(efp + 2);
    f[3] = __builtin_nontemporal_load(efp + 3);
    const float* fv = (const float*)f;

    float G = f[0].x, B = f[0].y;
    float Ire = G * Vre - B * Vim;
    float Iim = G * Vim + B * Vre;

    // edge_inputs = [Vre, Vim, ef[0..15], Ire, Iim] @ W_res(20x2) + b_res
    float p0 = b_res[0] + Vre * W_res[0] + Vim * W_res[2]
             + Ire * W_res[36] + Iim * W_res[38];
    float p1 = b_res[1] + Vre * W_res[1] + Vim * W_res[3]
             + Ire * W_res[37] + Iim * W_res[39];
#pragma unroll
    for (int k = 0; k < 16; ++k) {
        p0 += fv[k] * W_res[(2 + k) * 2 + 0];
        p1 += fv[k] * W_res[(2 + k) * 2 + 1];
    }

    float I0 = Ire + p0;
    float I1 = Iim + p1;

    // I_pred output (streaming write)
    __builtin_nontemporal_store(I0, I_pred + (size_t)e * 2 + 0);
    __builtin_nontemporal_store(I1, I_pred + (size_t)e * 2 + 1);

    // net_current = segsum(I_pred, receivers) - segsum(I_pred, senders)
    __hip_atomic_fetch_add(&net_cur[(size_t)r * 2 + 0],  I0, __ATOMIC_RELAXED, __HIP_MEMORY_SCOPE_AGENT);
    __hip_atomic_fetch_add(&net_cur[(size_t)r * 2 + 1],  I1, __ATOMIC_RELAXED, __HIP_MEMORY_SCOPE_AGENT);
    __hip_atomic_fetch_add(&net_cur[(size_t)s * 2 + 0], -I0, __ATOMIC_RELAXED, __HIP_MEMORY_SCOPE_AGENT);
    __hip_atomic_fetch_add(&net_cur[(size_t)s * 2 + 1], -I1, __ATOMIC_RELAXED, __HIP_MEMORY_SCOPE_AGENT);
}

// ---------------------------------------------------------------------------
// Kernel 2: node MLP  relu([V_node | net_current] @ W_node + b_node)
//   Full-fp32 WMMA: V_WMMA_F32_16X16X4_F32.
//   One wave handles a 16-node tile x all 64 output cols (4 accumulators).
//   K = 64 (V_node, 16 steps of K=4) + tail step [net0, net1, 0, 0].
// ---------------------------------------------------------------------------
__global__ __launch_bounds__(256) void node_kernel(
    const float* __restrict__ V_node,   // (N,64)
    const float* __restrict__ net_cur,  // (N,2)
    const float* __restrict__ W_node,   // (66,64) row-major
    const float* __restrict__ b_node,   // (64)
    float*       __restrict__ out,      // (N,64)
    int n_nodes)
{
    int wave  = (int)((blockIdx.x * blockDim.x + threadIdx.x) >> 5);
    int lane  = threadIdx.x & 31;
    int node0 = wave * 16;
    if (node0 >= n_nodes) return;   // wave-uniform: EXEC stays all-ones

    int half = lane >> 4;   // 0: lanes 0-15, 1: lanes 16-31
    int l    = lane & 15;
    int row  = node0 + l;   // A-matrix row M owned by this lane

    v8f acc[4] = {};

    const float* arow = V_node + (size_t)row * 64;

    // K = 0..63 over V_node columns
    for (int k0 = 0; k0 < 64; k0 += 4) {
        // A 16x4: lane<16 -> K={k0,k0+1}; lane>=16 -> K={k0+2,k0+3}
        v2f a = *(const v2f*)(arow + k0 + 2 * half);
#pragma unroll
        for (int t = 0; t < 4; ++t) {
            int n = t * 16 + l;  // output column
            // B 4x16: VGPR0 = rows {k0, k0+2}, VGPR1 = rows {k0+1, k0+3}
            v2f b;
            b.x = W_node[(size_t)(k0 + 2 * half + 0) * 64 + n];
            b.y = W_node[(size_t)(k0 + 2 * half + 1) * 64 + n];
            acc[t] = __builtin_amdgcn_wmma_f32_16x16x4_f32(
                false, a, false, b, (short)0, acc[t], false, false);
        }
    }

    // Tail: K = 64,65 carry net_current; K = 66,67 are zero padding.
    {
        float m = half ? 0.0f : 1.0f;   // zero upper-half K-slots, no divergence
        v2f nc = *(const v2f*)(net_cur + (size_t)row * 2);
        v2f a; a.x = m * nc.x; a.y = m * nc.y;
#pragma unroll
        for (int t = 0; t < 4; ++t) {
            int n = t * 16 + l;
            v2f b;
            b.x = m * W_node[(size_t)64 * 64 + n];
            b.y = m * W_node[(size_t)65 * 64 + n];
            acc[t] = __builtin_amdgcn_wmma_f32_16x16x4_f32(
                false, a, false, b, (short)0, acc[t], false, false);
        }
    }

    // Store with bias + ReLU. C/D layout: VGPR j -> M=j+8*half, N = t*16 + l.
    // For fixed (t,j) lanes 0..15 write 64 contiguous bytes -> coalesced.
#pragma unroll
    for (int t = 0; t < 4; ++t) {
        int n = t * 16 + l;
        float bn = b_node[n];
#pragma unroll
        for (int j = 0; j < 8; ++j) {
            int m = j + 8 * half;
            float v = acc[t][j] + bn;
            out[(size_t)(node0 + m) * 64 + n] = v > 0.0f ? v : 0.0f;
        }
    }
}

// ---------------------------------------------------------------------------
extern "C" void kernel_launch(void* const* d_in, const int* in_sizes, int n_in,
                              void* d_out, int out_size, void* d_ws, size_t ws_size,
                              hipStream_t stream) {
    const float* V_node    = (const float*)d_in[0];
    const int*   senders   = (const int*)  d_in[1];
    const int*   receivers = (const int*)  d_in[2];
    const float* ef        = (const float*)d_in[3];
    const float* W_res     = (const float*)d_in[4];
    const float* b_res     = (const float*)d_in[5];
    const float* W_node    = (const float*)d_in[6];
    const float* b_node    = (const float*)d_in[7];

    int n_nodes = in_sizes[0] / 64;   // 100000
    int n_edges = in_sizes[1];        // 3200000

    float* out_nodes = (float*)d_out;                     // (N,64)
    float* out_Ipred = out_nodes + (size_t)n_nodes * 64;  // (E,2)
    float* net_cur   = (float*)d_ws;                      // (N,2) scratch

    // 1) zero the segment-sum accumulator
    int nz = n_nodes * 2;
    zero_kernel<<<(nz + 255) / 256, 256, 0, stream>>>(net_cur, nz);

    // 2) edge phase: physics + residual + I_pred + scatter-add
    edge_kernel<<<(n_edges + 255) / 256, 256, 0, stream>>>(
        V_node, senders, receivers, ef, W_res, b_res,
        out_Ipred, net_cur, n_edges);

    // 3) node phase: WMMA f32 GEMM + bias + ReLU
    int nwaves  = (n_nodes + 15) / 16;       // 6250
    int nblocks = (nwaves + 7) / 8;          // 8 waves (256 thr) per block
    node_kernel<<<nblocks, 256, 0, stream>>>(
        V_node, net_cur, W_node, b_node, out_nodes, n_nodes);
}